// GCNEncoder_73675868995795
// MI455X (gfx1250) — compile-verified
//
#include <hip/hip_runtime.h>

#define NNODES 10000
#define EDGES  320000
#define INDIM  128
#define HID    256
#define NLAYER 6
#define ALPHA  0.1f
#define LNEPS  1e-5f

typedef __attribute__((ext_vector_type(2))) float v2f;
typedef __attribute__((ext_vector_type(8))) float v8f;

__device__ __forceinline__ float gelu_exact(float x) {
    return 0.5f * x * (1.0f + erff(x * 0.70710678118654752f));
}

// ---------------- degree / symmetric norm ----------------
__global__ void k_zero_u32(unsigned* p, int n) {
    int i = blockIdx.x * blockDim.x + threadIdx.x;
    if (i < n) p[i] = 0u;
}

__global__ void k_count_deg(const long long* __restrict__ dst,
                            unsigned* __restrict__ cnt, int e) {
    int i = blockIdx.x * blockDim.x + threadIdx.x;
    if (i < e) atomicAdd(&cnt[(int)dst[i]], 1u);
}

__global__ void k_dinv(const unsigned* __restrict__ cnt,
                       float* __restrict__ dinv, int n) {
    int i = blockIdx.x * blockDim.x + threadIdx.x;
    if (i < n) dinv[i] = rsqrtf((float)(cnt[i] + 1u)); // +1 self loop => always > 0
}

// ---------------- fp32 WMMA GEMM: C = A(MxK) * B(KxN) [+ bias] ----------------
#define BM 64
#define BN 64
#define KC 16

__global__ __launch_bounds__(256) void k_gemm_wmma(
    const float* __restrict__ A, const float* __restrict__ B,
    const float* __restrict__ bias, float* __restrict__ C,
    int M, int K, int N)
{
    __shared__ float As[BM][KC + 1];
    __shared__ float Bs[KC][BN + 1];

    const int tid  = threadIdx.x;
    const int wave = tid >> 5;
    const int lane = tid & 31;
    const int h    = lane >> 4;   // lane half
    const int lm   = lane & 15;
    const int tr   = wave & 3;            // tile row 0..3
    const int tc0  = (wave >> 2) << 1;    // tile col pair base: 0 or 2

    const int rowBase = blockIdx.x * BM;
    const int colBase = blockIdx.y * BN;

    v8f acc0 = {};
    v8f acc1 = {};

    const int arow = tid >> 2;          // 0..63
    const int akf  = (tid & 3) << 2;    // 0,4,8,12
    const int brow = tid >> 4;          // 0..15
    const int bcf  = (tid & 15) << 2;   // 0..60

    for (int kc = 0; kc < K; kc += KC) {
        // stage A 64x16 tile
        float4 av = make_float4(0.f, 0.f, 0.f, 0.f);
        int gr = rowBase + arow;
        if (gr < M) av = *(const float4*)(A + (size_t)gr * K + kc + akf);
        As[arow][akf + 0] = av.x; As[arow][akf + 1] = av.y;
        As[arow][akf + 2] = av.z; As[arow][akf + 3] = av.w;
        // stage B 16x64 tile (K,N multiples of 16/64 by construction)
        float4 bv = *(const float4*)(B + (size_t)(kc + brow) * N + colBase + bcf);
        Bs[brow][bcf + 0] = bv.x; Bs[brow][bcf + 1] = bv.y;
        Bs[brow][bcf + 2] = bv.z; Bs[brow][bcf + 3] = bv.w;
        __syncthreads();

        const int rA = tr * 16 + lm;
        const int c0 = tc0 * 16 + lm;
        const int c1 = c0 + 16;
#pragma unroll
        for (int k4 = 0; k4 < KC; k4 += 4) {
            const int k = k4 + 2 * h;          // A/B frag: K = 2*half + vgpr
            v2f a;  a.x  = As[rA][k];  a.y  = As[rA][k + 1];
            v2f b0; b0.x = Bs[k][c0];  b0.y = Bs[k + 1][c0];
            v2f b1; b1.x = Bs[k][c1];  b1.y = Bs[k + 1][c1];
            acc0 = __builtin_amdgcn_wmma_f32_16x16x4_f32(
                false, a, false, b0, (short)0, acc0, false, false);
            acc1 = __builtin_amdgcn_wmma_f32_16x16x4_f32(
                false, a, false, b1, (short)0, acc1, false, false);
        }
        __syncthreads();
    }

    // D layout: VGPR r holds M = r + 8*half, N = lane%16
    const int cg0 = colBase + tc0 * 16 + lm;
    const int cg1 = cg0 + 16;
    const float bb0 = bias ? bias[cg0] : 0.f;
    const float bb1 = bias ? bias[cg1] : 0.f;
#pragma unroll
    for (int r = 0; r < 8; ++r) {
        int gm = rowBase + tr * 16 + h * 8 + r;
        if (gm < M) {
            C[(size_t)gm * N + cg0] = acc0[r] + bb0;
            C[(size_t)gm * N + cg1] = acc1[r] + bb1;
        }
    }
}

// ---------------- GELU + LayerNorm helpers (one block of 256 per row) ----------------
__device__ __forceinline__ void block_meanvar(float g, float& mean, float& rstd) {
    float s1 = g, s2 = g * g;
#pragma unroll
    for (int off = 16; off > 0; off >>= 1) {
        s1 += __shfl_down(s1, off);
        s2 += __shfl_down(s2, off);
    }
    __shared__ float r1[8], r2[8], mv[2];
    const int wave = threadIdx.x >> 5, lane = threadIdx.x & 31;
    if (lane == 0) { r1[wave] = s1; r2[wave] = s2; }
    __syncthreads();
    if (threadIdx.x == 0) {
        float a = 0.f, b = 0.f;
#pragma unroll
        for (int i = 0; i < 8; ++i) { a += r1[i]; b += r2[i]; }
        float mu = a * (1.0f / HID);
        mv[0] = mu;
        mv[1] = rsqrtf(b * (1.0f / HID) - mu * mu + LNEPS);
    }
    __syncthreads();
    mean = mv[0];
    rstd = mv[1];
}

__global__ __launch_bounds__(HID) void k_gelu_ln_in(
    const float* __restrict__ t, const float* __restrict__ gamma,
    const float* __restrict__ beta, float* __restrict__ h0, float* __restrict__ cur)
{
    const int row = blockIdx.x, c = threadIdx.x;
    const size_t idx = (size_t)row * HID + c;
    float g = gelu_exact(t[idx]);
    float mean, rstd;
    block_meanvar(g, mean, rstd);
    float o = (g - mean) * rstd * gamma[c] + beta[c];
    h0[idx]  = o;
    cur[idx] = o;
}

__global__ __launch_bounds__(HID) void k_post(
    const float* __restrict__ agg, const float* __restrict__ bl,
    const float* __restrict__ gl, const float* __restrict__ betal,
    const float* __restrict__ cur_in, const float* __restrict__ h0,
    float* __restrict__ cur_out)
{
    const int row = blockIdx.x, c = threadIdx.x;
    const size_t idx = (size_t)row * HID + c;
    float g = gelu_exact(agg[idx] + bl[c]);
    float mean, rstd;
    block_meanvar(g, mean, rstd);
    float hh = (g - mean) * rstd * gl[c] + betal[c];
    cur_out[idx] = cur_in[idx] + (1.0f - ALPHA) * hh + ALPHA * h0[idx];
}

// ---------------- message passing ----------------
// self-loop term doubles as agg initialization: agg = dinv^2 * m
__global__ __launch_bounds__(HID) void k_self(
    const float* __restrict__ m, const float* __restrict__ dinv,
    float* __restrict__ agg)
{
    const int row = blockIdx.x, c = threadIdx.x;
    float dv = dinv[row];
    agg[(size_t)row * HID + c] = dv * dv * m[(size_t)row * HID + c];
}

// one wave per edge; 32 lanes x 8 coalesced channels
__global__ __launch_bounds__(256) void k_scatter(
    const long long* __restrict__ src, const long long* __restrict__ dst,
    const float* __restrict__ dinv, const float* __restrict__ m,
    float* __restrict__ agg, int e)
{
    const int wave = threadIdx.x >> 5;
    const int lane = threadIdx.x & 31;
    const int ei = blockIdx.x * 8 + wave;
    if (ei >= e) return;
    const int s = (int)src[ei];
    const int d = (int)dst[ei];
    const float nrm = dinv[s] * dinv[d];
    const float* mp = m + (size_t)s * HID;
    float* ap = agg + (size_t)d * HID;
#pragma unroll
    for (int j = 0; j < HID / 32; ++j) {
        int cidx = lane + 32 * j;
        atomicAdd(&ap[cidx], nrm * mp[cidx]);
    }
}

// ---------------- host orchestration ----------------
extern "C" void kernel_launch(void* const* d_in, const int* in_sizes, int n_in,
                              void* d_out, int out_size, void* d_ws, size_t ws_size,
                              hipStream_t stream)
{
    const float*     x     = (const float*)d_in[0];
    const long long* ei    = (const long long*)d_in[1];
    const float*     W_in  = (const float*)d_in[2];
    const float*     b_in  = (const float*)d_in[3];
    const float*     g_in  = (const float*)d_in[4];
    const float*     be_in = (const float*)d_in[5];
    const float*     Wl    = (const float*)d_in[6];
    const float*     bl    = (const float*)d_in[7];
    const float*     gl    = (const float*)d_in[8];
    const float*     betal = (const float*)d_in[9];
    float* out = (float*)d_out;

    const long long* esrc = ei;
    const long long* edst = ei + EDGES;

    char* ws = (char*)d_ws;
    const size_t fmat = (size_t)NNODES * HID;
    float* mbuf   = (float*)ws; ws += fmat * sizeof(float);
    float* h0     = (float*)ws; ws += fmat * sizeof(float);
    float* cur    = (float*)ws; ws += fmat * sizeof(float);
    float* agg    = (float*)ws; ws += fmat * sizeof(float);
    float* dinv   = (float*)ws; ws += NNODES * sizeof(float);
    unsigned* cnt = (unsigned*)ws;

    // degree + normalization
    k_zero_u32<<<(NNODES + 255) / 256, 256, 0, stream>>>(cnt, NNODES);
    k_count_deg<<<(EDGES + 255) / 256, 256, 0, stream>>>(edst, cnt, EDGES);
    k_dinv<<<(NNODES + 255) / 256, 256, 0, stream>>>(cnt, dinv, NNODES);

    // input block: Linear(+bias) -> GELU -> LayerNorm
    dim3 gIn((NNODES + BM - 1) / BM, HID / BN);
    k_gemm_wmma<<<gIn, 256, 0, stream>>>(x, W_in, b_in, mbuf, NNODES, INDIM, HID);
    k_gelu_ln_in<<<NNODES, HID, 0, stream>>>(mbuf, g_in, be_in, h0, cur);

    dim3 gL((NNODES + BM - 1) / BM, HID / BN);
    for (int l = 0; l < NLAYER; ++l) {
        k_gemm_wmma<<<gL, 256, 0, stream>>>(
            cur, Wl + (size_t)l * HID * HID, nullptr, mbuf, NNODES, HID, HID);
        k_self<<<NNODES, HID, 0, stream>>>(mbuf, dinv, agg);
        k_scatter<<<(EDGES + 7) / 8, 256, 0, stream>>>(esrc, edst, dinv, mbuf, agg, EDGES);
        float* co = (l == NLAYER - 1) ? out : cur;
        k_post<<<NNODES, HID, 0, stream>>>(
            agg, bl + l * HID, gl + l * HID, betal + l * HID, cur, h0, co);
    }
}